// RearrangeLinear2_89919435309635
// MI455X (gfx1250) — compile-verified
//
#include <hip/hip_runtime.h>
#include <stdint.h>

#define TPB 256      // threads per block in the gather kernel
#define RPB 4        // output rows per block (R % RPB == 0, rows stay in one batch)

// ---------------------------------------------------------------------------
// CDNA5 async global->LDS helpers (gfx1250 only; tracked by ASYNCcnt)
// ---------------------------------------------------------------------------
__device__ __forceinline__ uint32_t lds_addr_of(const void* p) {
  // addrspacecast(LDS -> generic) puts the LDS byte offset in the low 32 bits.
  return (uint32_t)(uintptr_t)p;
}

__device__ __forceinline__ void async_load_b128(uint32_t lds_byte_addr, uint64_t gaddr) {
  asm volatile("global_load_async_to_lds_b128 %0, %1, off"
               :
               : "v"(lds_byte_addr), "v"(gaddr)
               : "memory");
}

__device__ __forceinline__ void wait_asynccnt0() {
  asm volatile("s_wait_asynccnt 0" ::: "memory");
}

// ---------------------------------------------------------------------------
// Kernel 1: build constrained permutation from a weight vector (one workgroup).
//   sel[i]  = round(clip(w[i], 0, n-1))                 (round-half-even)
//   dup[i]  = sel[i] seen at an earlier index
//   k-th duplicate (index order) -> k-th LARGEST unused value
// ---------------------------------------------------------------------------
__global__ __launch_bounds__(1024)
void build_perm_kernel(const float* __restrict__ w, int* __restrict__ perm, int n) {
  __shared__ int s_sel[2048];
  __shared__ int s_first[2048];
  __shared__ int s_dup[2048];
  __shared__ int s_a[2048];
  __shared__ int s_b[2048];
  __shared__ int s_empty[2048];
  const int T   = blockDim.x;
  const int tid = threadIdx.x;

  for (int i = tid; i < n; i += T) {
    float v = fminf(fmaxf(w[i], 0.0f), (float)(n - 1));
    s_sel[i]   = (int)rintf(v);        // matches jnp.round (RNE)
    s_first[i] = 0x7fffffff;
  }
  __syncthreads();
  for (int i = tid; i < n; i += T) atomicMin(&s_first[s_sel[i]], i);
  __syncthreads();
  for (int i = tid; i < n; i += T) {
    int d = (s_first[s_sel[i]] != i) ? 1 : 0;
    s_dup[i] = d;
    s_a[i]   = d;
  }
  __syncthreads();

  // scan #1: inclusive scan of dup flags -> duplicate rank (Hillis-Steele)
  int* p = s_a; int* q = s_b;
  for (int off = 1; off < n; off <<= 1) {
    for (int i = tid; i < n; i += T)
      q[i] = (i >= off) ? (p[i] + p[i - off]) : p[i];
    __syncthreads();
    int* t = p; p = q; q = t;
  }
  int myrank[2];                                   // n/T <= 2 with 1024 threads
  for (int i = tid, k = 0; i < n; i += T, ++k) myrank[k] = p[i] - 1;
  __syncthreads();

  // scan #2: over reversed unused-flags -> descending rank of unused values
  for (int i = tid; i < n; i += T) {
    int v = n - 1 - i;
    s_a[i] = (s_first[v] == 0x7fffffff) ? 1 : 0;
  }
  __syncthreads();
  p = s_a; q = s_b;
  for (int off = 1; off < n; off <<= 1) {
    for (int i = tid; i < n; i += T)
      q[i] = (i >= off) ? (p[i] + p[i - off]) : p[i];
    __syncthreads();
    int* t = p; p = q; q = t;
  }
  // scatter: k-th largest unused value -> s_empty[k]
  for (int v = tid; v < n; v += T)
    if (s_first[v] == 0x7fffffff) s_empty[p[n - 1 - v] - 1] = v;
  __syncthreads();

  for (int i = tid, k = 0; i < n; i += T, ++k)
    perm[i] = s_dup[i] ? s_empty[myrank[k]] : s_sel[i];
}

// ---------------------------------------------------------------------------
// Kernel 2: out[batch, i, j] = x[batch, r2[i], b2[j]]
// Async-copy RPB source rows + column perm into LDS, gather from LDS,
// write coalesced float4 stores. All HBM traffic is fully coalesced.
// ---------------------------------------------------------------------------
__global__ __launch_bounds__(TPB)
void gather_kernel(const float* __restrict__ x,
                   const int*   __restrict__ colperm,
                   const int*   __restrict__ rowperm,
                   float*       __restrict__ out,
                   int R, int C) {
  __shared__ __align__(16) float tile[RPB * 2048];
  __shared__ __align__(16) int   s_col[2048];
  const int tid    = threadIdx.x;
  const int row0   = blockIdx.x * RPB;   // flat output row (batch*R + i)
  const int batch  = row0 / R;
  const int chunks = C / 4;              // 16-byte chunks per C ints / C floats

  // stage column permutation (C ints)
  {
    const uint32_t l0 = lds_addr_of(s_col);
    const uint64_t g0 = (uint64_t)(uintptr_t)colperm;
    for (int c = tid; c < chunks; c += TPB)
      async_load_b128(l0 + (uint32_t)c * 16u, g0 + (uint64_t)c * 16u);
  }
  // stage RPB source rows of x
  for (int k = 0; k < RPB; ++k) {
    const int srow    = rowperm[row0 + k - batch * R];
    const uint64_t g0 = (uint64_t)(uintptr_t)(x + ((size_t)batch * R + srow) * C);
    const uint32_t l0 = lds_addr_of(tile + k * C);
    for (int c = tid; c < chunks; c += TPB)
      async_load_b128(l0 + (uint32_t)c * 16u, g0 + (uint64_t)c * 16u);
  }
  wait_asynccnt0();
  __syncthreads();

  for (int k = 0; k < RPB; ++k) {
    const float* t = tile + k * C;
    float*       o = out + (size_t)(row0 + k) * C;
    for (int j = tid * 4; j < C; j += TPB * 4) {
      float4 v;
      v.x = t[s_col[j + 0]];
      v.y = t[s_col[j + 1]];
      v.z = t[s_col[j + 2]];
      v.w = t[s_col[j + 3]];
      *reinterpret_cast<float4*>(o + j) = v;
    }
  }
}

// ---------------------------------------------------------------------------
extern "C" void kernel_launch(void* const* d_in, const int* in_sizes, int n_in,
                              void* d_out, int out_size, void* d_ws, size_t ws_size,
                              hipStream_t stream) {
  const float* x  = (const float*)d_in[0];
  const float* wc = (const float*)d_in[1];
  const float* wr = (const float*)d_in[2];
  float* out = (float*)d_out;

  const int C = in_sizes[1];             // 2048
  const int R = in_sizes[2];             // 1024
  const int B = in_sizes[0] / (R * C);   // 16

  int* colperm = (int*)d_ws;             // C ints
  int* rowperm = colperm + C;            // R ints

  build_perm_kernel<<<1, 1024, 0, stream>>>(wc, colperm, C);
  build_perm_kernel<<<1, 1024, 0, stream>>>(wr, rowperm, R);
  gather_kernel<<<(B * R) / RPB, TPB, 0, stream>>>(x, colperm, rowperm, out, R, C);
}